// TopKPooling_11776800326279
// MI455X (gfx1250) — compile-verified
//
#include <hip/hip_runtime.h>
#include <hip/hip_bf16.h>

typedef __attribute__((ext_vector_type(2))) float v2f;
typedef __attribute__((ext_vector_type(8))) float v8f;

#define NUM_BATCH 64
#define NPB       4096
#define NTOT      (NUM_BATCH * NPB)     /* 262144 nodes   */
#define CCH       128                   /* channels       */
#define KSEL      2048                  /* ratio 0.5      */
#define PSEL      (NUM_BATCH * KSEL)    /* 131072 kept    */
#define EDG       (32 * NTOT)           /* 8388608 edges  */

// ---------------------------------------------------------------------------
// K1: score[m] = dot(x[m,:], p)  via V_WMMA_F32_16X16X4_F32.
// One wave computes 16 rows. A tile = x[m0..m0+15, k0..k0+3] (f32 A layout:
// lane l holds M=l%16, K=k0+2*(l/16)+{0,1}). B = p[k0+K] broadcast over all
// 16 columns, so every column of D accumulates the same dot product.
// D layout: lane l, VGPR v <-> row m0 + v + 8*(l/16); read column 0.
// ---------------------------------------------------------------------------
__global__ __launch_bounds__(256) void score_kernel(const float* __restrict__ x,
                                                    const float* __restrict__ p,
                                                    float* __restrict__ score) {
    const int lane  = threadIdx.x & 31;
    const int gwave = (int)((blockIdx.x * blockDim.x + threadIdx.x) >> 5);
    const int m0    = gwave * 16;                 // grid sized exactly: no bounds check
    const int half  = (lane >> 4) & 1;
    const float* __restrict__ xr = x + (size_t)(m0 + (lane & 15)) * CCH + half * 2;
    const float* __restrict__ pr = p + half * 2;

    v8f c = {};
#pragma unroll
    for (int k0 = 0; k0 < CCH; k0 += 4) {
        v2f a, b;
        a[0] = xr[k0];
        a[1] = xr[k0 + 1];
        b[0] = pr[k0];
        b[1] = pr[k0 + 1];
        c = __builtin_amdgcn_wmma_f32_16x16x4_f32(
                /*neg_a=*/false, a, /*neg_b=*/false, b,
                /*c_mod=*/(short)0, c, /*reuse_a=*/false, /*reuse_b=*/false);
    }
    if ((lane & 15) == 0) {
#pragma unroll
        for (int v = 0; v < 8; ++v)
            score[m0 + half * 8 + v] = c[v];
    }
}

// ---------------------------------------------------------------------------
// K2: per-batch (contiguous 4096-node segment) softmax, in place.
// ---------------------------------------------------------------------------
__global__ __launch_bounds__(1024) void softmax_kernel(float* __restrict__ score) {
    __shared__ float red[1024];
    const int tid = threadIdx.x;
    float* __restrict__ s = score + (size_t)blockIdx.x * NPB;

    float v0 = s[tid], v1 = s[tid + 1024], v2 = s[tid + 2048], v3 = s[tid + 3072];

    red[tid] = fmaxf(fmaxf(v0, v1), fmaxf(v2, v3));
    __syncthreads();
    for (int off = 512; off > 0; off >>= 1) {
        if (tid < off) red[tid] = fmaxf(red[tid], red[tid + off]);
        __syncthreads();
    }
    const float mx = red[0];
    __syncthreads();

    float e0 = expf(v0 - mx), e1 = expf(v1 - mx), e2 = expf(v2 - mx), e3 = expf(v3 - mx);
    red[tid] = (e0 + e1) + (e2 + e3);
    __syncthreads();
    for (int off = 512; off > 0; off >>= 1) {
        if (tid < off) red[tid] += red[tid + off];
        __syncthreads();
    }
    const float inv = 1.0f / red[0];

    s[tid]        = e0 * inv;
    s[tid + 1024] = e1 * inv;
    s[tid + 2048] = e2 * inv;
    s[tid + 3072] = e3 * inv;
}

// ---------------------------------------------------------------------------
// K3: bitonic argsort (ascending, (key,index) lexicographic == stable sort)
// of batch b's 4096 scores in LDS. Output row r = 63-b (the [::-1] reverses
// the batch axis in the reference): perm = sorted local index + r*4096,
// take first KSEL. Also scatter node_mask (this block owns region r) and
// write batch_out[q] = min(perm[q], PSEL-1) per JAX's clamped OOB gather.
// ---------------------------------------------------------------------------
__global__ __launch_bounds__(1024) void sort_kernel(const float* __restrict__ score,
                                                    int* __restrict__ perm,
                                                    int* __restrict__ node_mask,
                                                    float* __restrict__ batch_out) {
    __shared__ float skey[NPB];
    __shared__ int   sidx[NPB];
    const int b   = blockIdx.x;
    const int tid = threadIdx.x;
    const float* __restrict__ s = score + (size_t)b * NPB;

    for (int i = tid; i < NPB; i += 1024) { skey[i] = s[i]; sidx[i] = i; }
    __syncthreads();

    for (int k = 2; k <= NPB; k <<= 1) {
        for (int j = k >> 1; j > 0; j >>= 1) {
            for (int i = tid; i < NPB; i += 1024) {
                const int ixj = i ^ j;
                if (ixj > i) {
                    const bool  up   = ((i & k) == 0);
                    const float ka   = skey[i], kb = skey[ixj];
                    const int   ia   = sidx[i], ib = sidx[ixj];
                    const bool  agtb = (ka > kb) || ((ka == kb) && (ia > ib));
                    if (agtb == up) {
                        skey[i] = kb; skey[ixj] = ka;
                        sidx[i] = ib; sidx[ixj] = ia;
                    }
                }
            }
            __syncthreads();
        }
    }

    const int r = (NUM_BATCH - 1) - b;
    // init node_mask for the contiguous region this block's output row targets
    for (int i = tid; i < NPB; i += 1024) node_mask[r * NPB + i] = -1;
    __syncthreads();

    for (int j = tid; j < KSEL; j += 1024) {
        const int g = r * NPB + sidx[j];       // global perm value
        const int q = r * KSEL + j;            // cluster id
        perm[q]      = g;
        node_mask[g] = q;
        batch_out[q] = (float)(g < PSEL ? g : (PSEL - 1));
    }
}

// ---------------------------------------------------------------------------
// K4: x_out[q,:] = x[perm[q],:] * score[perm[q]]   (one wave per row, float4)
// ---------------------------------------------------------------------------
__global__ __launch_bounds__(256) void gather_kernel(const float* __restrict__ x,
                                                     const float* __restrict__ score,
                                                     const int* __restrict__ perm,
                                                     float* __restrict__ xout) {
    const int lane = threadIdx.x & 31;
    const int q    = (int)((blockIdx.x * blockDim.x + threadIdx.x) >> 5);
    const int g    = perm[q];
    const float sc = score[g];
    const float4 v = *(const float4*)(x + (size_t)g * CCH + lane * 4);
    float4 o;
    o.x = v.x * sc; o.y = v.y * sc; o.z = v.z * sc; o.w = v.w * sc;
    *(float4*)(xout + (size_t)q * CCH + lane * 4) = o;
}

// ---------------------------------------------------------------------------
// K5: edge remap through node_mask, masked to -1.
// ---------------------------------------------------------------------------
__global__ __launch_bounds__(256) void edge_kernel(const int* __restrict__ senders,
                                                   const int* __restrict__ receivers,
                                                   const int* __restrict__ node_mask,
                                                   float* __restrict__ s_out,
                                                   float* __restrict__ r_out) {
    const int e  = (int)(blockIdx.x * blockDim.x + threadIdx.x);
    const int s2 = node_mask[senders[e]];
    const int r2 = node_mask[receivers[e]];
    const bool m = (s2 >= 0) && (r2 >= 0);
    s_out[e] = (float)(m ? s2 : -1);
    r_out[e] = (float)(m ? r2 : -1);
}

// ---------------------------------------------------------------------------
extern "C" void kernel_launch(void* const* d_in, const int* in_sizes, int n_in,
                              void* d_out, int out_size, void* d_ws, size_t ws_size,
                              hipStream_t stream) {
    const float* x         = (const float*)d_in[0];
    const float* p         = (const float*)d_in[1];
    const int*   senders   = (const int*)d_in[2];
    const int*   receivers = (const int*)d_in[3];
    // d_in[4] (batch) is implied by the contiguous uniform segments.

    float* out   = (float*)d_out;
    float* xout  = out;                                   // PSEL*CCH
    float* s_out = out + (size_t)PSEL * CCH;              // EDG
    float* r_out = s_out + (size_t)EDG;                   // EDG
    float* b_out = r_out + (size_t)EDG;                   // PSEL

    float* score     = (float*)d_ws;                      // NTOT floats
    int*   perm      = (int*)(score + NTOT);              // PSEL ints
    int*   node_mask = perm + PSEL;                       // NTOT ints

    // 16384 row-tiles of 16, 8 waves/block -> 2048 blocks
    score_kernel<<<NTOT / 16 / 8, 256, 0, stream>>>(x, p, score);
    softmax_kernel<<<NUM_BATCH, 1024, 0, stream>>>(score);
    sort_kernel<<<NUM_BATCH, 1024, 0, stream>>>(score, perm, node_mask, b_out);
    gather_kernel<<<PSEL / 8, 256, 0, stream>>>(x, score, perm, xout);
    edge_kernel<<<EDG / 256, 256, 0, stream>>>(senders, receivers, node_mask, s_out, r_out);
}